// GraphInteractionLayer_31224412242367
// MI455X (gfx1250) — compile-verified
//
#include <hip/hip_runtime.h>

typedef float v2f __attribute__((ext_vector_type(2)));
typedef float v8f __attribute__((ext_vector_type(8)));

#define BB     8
#define NN     256
#define DNODE  64
#define DEDGE  32
#define DCODE  64
#define DOUT   60   // D_NODE - 4

// ---------------------------------------------------------------------------
// Kernel 1: tiny precompute GEMMs into workspace.
//   rc[b,n,c]   = node_info[b,n,:] . W_e[32+k, c]              (receiver term)
//   sc[b,n,c]   = node_info[b,n,:] . W_e[96+k, c] + b_e[c]     (sender + bias)
//   base[b,n,o] = node_info[b,n,:] . W_n[k, o]    + b_n[o]     (node partial)
// b_e is folded into sc so the hot kernel's WMMA C-init covers all biases.
// ---------------------------------------------------------------------------
__global__ void gil_pre(const float* __restrict__ node_info,
                        const float* __restrict__ W_e,
                        const float* __restrict__ b_e,
                        const float* __restrict__ W_n,
                        const float* __restrict__ b_n,
                        float* __restrict__ rc,
                        float* __restrict__ sc,
                        float* __restrict__ base)
{
    const int n_rc   = BB * NN * DCODE;
    const int n_base = BB * NN * DOUT;
    int idx = blockIdx.x * blockDim.x + threadIdx.x;
    if (idx < n_rc) {
        int row = idx / DCODE, c = idx % DCODE;
        const float* ni = node_info + (size_t)row * DNODE;
        float a = 0.f;
        #pragma unroll 8
        for (int k = 0; k < DNODE; ++k)
            a += ni[k] * W_e[(DEDGE + k) * DCODE + c];
        rc[idx] = a;
    } else if (idx < 2 * n_rc) {
        int j = idx - n_rc;
        int row = j / DCODE, c = j % DCODE;
        const float* ni = node_info + (size_t)row * DNODE;
        float a = b_e[c];
        #pragma unroll 8
        for (int k = 0; k < DNODE; ++k)
            a += ni[k] * W_e[(DEDGE + DNODE + k) * DCODE + c];
        sc[j] = a;
    } else if (idx < 2 * n_rc + n_base) {
        int j = idx - 2 * n_rc;
        int row = j / DOUT, o = j % DOUT;
        const float* ni = node_info + (size_t)row * DNODE;
        float a = b_n[o];
        #pragma unroll 8
        for (int k = 0; k < DNODE; ++k)
            a += ni[k] * W_n[k * DOUT + o];
        base[j] = a;
    }
}

// ---------------------------------------------------------------------------
// Kernel 2: heavy fused kernel. One block per (batch b, 16-receiver tile).
// 256 threads = 8 waves; wave w owns receivers r0+w and r0+w+8.
// Per 16-sender tile: D = edge_attr_tile @ W_e[0:32] + C, with C initialized
// to rc + (sc + b_e) so the biases ride through the WMMA accumulator.
// Then leaky_relu = max(x, 0.01x), multiply by adj, reduce over senders.
// Final: out = base + edge_agg @ W_n[64:128] from LDS.
// ---------------------------------------------------------------------------
__global__ __launch_bounds__(256)
void gil_edge(const float* __restrict__ edge_attr,
              const float* __restrict__ adj,
              const float* __restrict__ W_e,
              const float* __restrict__ W_n,
              const float* __restrict__ rc,
              const float* __restrict__ sc,
              const float* __restrict__ base,
              float* __restrict__ out)
{
    __shared__ float agg[16 * DCODE];   // edge_agg for this receiver tile

    const int b    = blockIdx.x >> 4;
    const int r0   = (blockIdx.x & 15) << 4;
    const int tid  = threadIdx.x;
    const int wave = tid >> 5;
    const int lane = tid & 31;
    const int lm   = lane & 15;   // M/N index within tile
    const int lh   = lane >> 4;   // lane-half selects K (A/B) or M-half (C/D)

    // B-operand fragments: W_e rows 0..31 (K) x 64 cols (N).
    // Layout: VGPR j, lane L -> K = j + 2*(L>=16), N = L%16.
    v2f wf[8][4];
    #pragma unroll
    for (int kk = 0; kk < 8; ++kk)
        #pragma unroll
        for (int ct = 0; ct < 4; ++ct) {
            int col = ct * 16 + lm;
            wf[kk][ct].x = W_e[(kk * 4 + 2 * lh    ) * DCODE + col];
            wf[kk][ct].y = W_e[(kk * 4 + 2 * lh + 1) * DCODE + col];
        }

    float rcr[2][4];
    #pragma unroll
    for (int ri = 0; ri < 2; ++ri)
        #pragma unroll
        for (int ct = 0; ct < 4; ++ct)
            rcr[ri][ct] = rc[((size_t)b * NN + r0 + wave + ri * 8) * DCODE + ct * 16 + lm];

    float acc[2][4];
    #pragma unroll
    for (int ri = 0; ri < 2; ++ri)
        #pragma unroll
        for (int ct = 0; ct < 4; ++ct) acc[ri][ct] = 0.f;

    for (int s0 = 0; s0 < NN; s0 += 16) {
        // sender-term fragments matching the D layout:
        // D VGPR v, lane L -> s_local = v + 8*(L>=16), c = ct*16 + L%16
        float scf[8][4];
        #pragma unroll
        for (int v = 0; v < 8; ++v) {
            const float* sp = sc + ((size_t)b * NN + s0 + v + 8 * lh) * DCODE;
            #pragma unroll
            for (int ct = 0; ct < 4; ++ct) scf[v][ct] = sp[ct * 16 + lm];
        }

        #pragma unroll
        for (int ri = 0; ri < 2; ++ri) {
            const int r = r0 + wave + ri * 8;
            const float* abase =
                edge_attr + (((size_t)b * NN + r) * NN + s0) * DEDGE;

            // prefetch next sender tile of this receiver's edge_attr rows
            if (s0 + 16 < NN)
                __builtin_prefetch(abase + (16 + lm) * DEDGE + lh * 16, 0, 0);

            // A-operand fragments: VGPR j, lane L -> M = L%16 (sender),
            // K = j + 2*(L>=16) -> contiguous float2 per lane.
            v2f af[8];
            #pragma unroll
            for (int kk = 0; kk < 8; ++kk)
                af[kk] = *(const v2f*)(abase + (size_t)lm * DEDGE + kk * 4 + 2 * lh);

            float adjv[8];
            const float* ap = adj + ((size_t)b * NN + r) * NN + s0;
            #pragma unroll
            for (int v = 0; v < 8; ++v) adjv[v] = ap[v + 8 * lh];

            #pragma unroll
            for (int ct = 0; ct < 4; ++ct) {
                // C-init carries rc + (sc + b_e): biases for free via D=A*B+C
                v8f d;
                #pragma unroll
                for (int i = 0; i < 8; ++i) d[i] = scf[i][ct] + rcr[ri][ct];

                #pragma unroll
                for (int kk = 0; kk < 8; ++kk)
                    d = __builtin_amdgcn_wmma_f32_16x16x4_f32(
                            /*neg_a=*/false, af[kk],
                            /*neg_b=*/false, wf[kk][ct],
                            /*c_mod=*/(short)0, d,
                            /*reuse_a=*/false, /*reuse_b=*/false);

                // leaky_relu (slope 0.01 < 1 => max form is exact),
                // adjacency mask, reduce over this half's 8 senders
                float part = 0.f;
                #pragma unroll
                for (int i = 0; i < 8; ++i) {
                    float x = d[i];
                    x = fmaxf(x, 0.01f * x);
                    part = fmaf(x, adjv[i], part);
                }
                // combine the two lane-halves (s_local 0..7 vs 8..15)
                part += __shfl_xor(part, 16, 32);
                acc[ri][ct] += part;
            }
        }
    }

    // stash per-receiver aggregates in LDS (value replicated in both halves)
    if (lh == 0) {
        #pragma unroll
        for (int ri = 0; ri < 2; ++ri)
            #pragma unroll
            for (int ct = 0; ct < 4; ++ct)
                agg[(wave + ri * 8) * DCODE + ct * 16 + lm] = acc[ri][ct];
    }
    __syncthreads();

    // final: out[b, r0+rl, o] = base + agg[rl,:] . W_n[64:128, o]
    for (int idx = tid; idx < 16 * DOUT; idx += 256) {
        int rl = idx / DOUT, o = idx % DOUT;
        float v = base[((size_t)b * NN + r0 + rl) * DOUT + o];
        #pragma unroll 8
        for (int c = 0; c < DCODE; ++c)
            v += agg[rl * DCODE + c] * W_n[(DNODE + c) * DOUT + o];
        out[((size_t)b * NN + r0 + rl) * DOUT + o] = v;
    }
}

// ---------------------------------------------------------------------------
extern "C" void kernel_launch(void* const* d_in, const int* in_sizes, int n_in,
                              void* d_out, int out_size, void* d_ws, size_t ws_size,
                              hipStream_t stream)
{
    const float* node_info = (const float*)d_in[0];
    const float* edge_attr = (const float*)d_in[1];
    const float* adj       = (const float*)d_in[2];
    const float* W_e       = (const float*)d_in[3];
    const float* b_e       = (const float*)d_in[4];
    const float* W_n       = (const float*)d_in[5];
    const float* b_n       = (const float*)d_in[6];
    float* out = (float*)d_out;

    // workspace layout (floats): rc | sc | base  (~1.5 MB total)
    float* rc   = (float*)d_ws;
    float* sc   = rc + BB * NN * DCODE;
    float* base = sc + BB * NN * DCODE;

    const int total_pre = 2 * BB * NN * DCODE + BB * NN * DOUT;
    gil_pre<<<(total_pre + 255) / 256, 256, 0, stream>>>(
        node_info, W_e, b_e, W_n, b_n, rc, sc, base);

    gil_edge<<<BB * (NN / 16), 256, 0, stream>>>(
        edge_attr, adj, W_e, W_n, rc, sc, base, out);
}